// TransitionDown_40613210751456
// MI455X (gfx1250) — compile-verified
//
#include <hip/hip_runtime.h>
#include <hip/hip_bf16.h>
#include <math.h>

// Problem constants (from the reference)
#define N_POINTS    16384
#define CHANNELS    256
#define NUM_SAMPLES 2048
#define KNN_K       16
#define MK          (NUM_SAMPLES * KNN_K)   // 32768 rows through the MLP
#define BN_EPS      1e-5f
#define FLT_BIG     3.402823466e38f

typedef __attribute__((ext_vector_type(2))) float v2f;
typedef __attribute__((ext_vector_type(8))) float v8f;

// ---------------------------------------------------------------------------
// Kernel 1: Farthest Point Sampling. One 1024-thread workgroup.
// Points cached in LDS (192KB of the 320KB CDNA5 WGP LDS); running min-dist
// kept in VGPRs (16 per thread). wave32 shuffle argmax + 2 barriers/iter.
// Tie-break: strictly-greater within a thread, (val desc, idx asc) across
// lanes => first-occurrence semantics, matching jnp.argmax.
// ---------------------------------------------------------------------------
__global__ void fps_kernel(const float* __restrict__ points,
                           int* __restrict__ sidx,
                           float* __restrict__ sampled_points) {
    extern __shared__ float sm[];
    float* px = sm;
    float* py = sm + N_POINTS;
    float* pz = sm + 2 * N_POINTS;
    __shared__ float rv[32];
    __shared__ int   ri[32];
    __shared__ int   s_far;

    const int t = threadIdx.x;          // 0..1023
    for (int i = t; i < N_POINTS; i += 1024) {
        px[i] = points[3 * i + 0];
        py[i] = points[3 * i + 1];
        pz[i] = points[3 * i + 2];
    }
    __syncthreads();

    float dj[16];
#pragma unroll
    for (int j = 0; j < 16; ++j) dj[j] = 1e10f;

    int far = 0;
    const int lane = t & 31, wv = t >> 5;

    for (int s = 0; s < NUM_SAMPLES; ++s) {
        if (t == 0) {
            sidx[s] = far;
            sampled_points[3 * s + 0] = px[far];
            sampled_points[3 * s + 1] = py[far];
            sampled_points[3 * s + 2] = pz[far];
        }
        const float cx = px[far], cy = py[far], cz = pz[far];

        float bestv = -1.0f; int besti = 0;
#pragma unroll
        for (int j = 0; j < 16; ++j) {
            const int idx = t + 1024 * j;
            const float dx = px[idx] - cx, dy = py[idx] - cy, dz = pz[idx] - cz;
            const float d = dx * dx + dy * dy + dz * dz;
            float dd = dj[j];
            dd = fminf(dd, d);
            dj[j] = dd;
            if (dd > bestv) { bestv = dd; besti = idx; }
        }
        // wave32 argmax reduce
        for (int off = 16; off > 0; off >>= 1) {
            const float ov = __shfl_xor(bestv, off, 32);
            const int   oi = __shfl_xor(besti, off, 32);
            if (ov > bestv || (ov == bestv && oi < besti)) { bestv = ov; besti = oi; }
        }
        if (lane == 0) { rv[wv] = bestv; ri[wv] = besti; }
        __syncthreads();
        if (t == 0) {
            float bv = rv[0]; int bi = ri[0];
            for (int w = 1; w < 32; ++w)
                if (rv[w] > bv || (rv[w] == bv && ri[w] < bi)) { bv = rv[w]; bi = ri[w]; }
            s_far = bi;
        }
        __syncthreads();
        far = s_far;
    }
}

// ---------------------------------------------------------------------------
// Kernel 2: 16-NN per sampled point. One 256-thread block per sample.
// Each thread keeps an unsorted register top-16 over 64 strided candidates
// (static indexing via per-slot selects), then 16 rounds of parallel argmin
// over the 4096 LDS candidates. Output order is irrelevant downstream
// (max-pool & batch-stats are K-permutation invariant).
// ---------------------------------------------------------------------------
__global__ void knn_kernel(const float* __restrict__ points,
                           const int* __restrict__ sidx,
                           int* __restrict__ knn_idx) {
    __shared__ float cd[256 * KNN_K];
    __shared__ int   ci[256 * KNN_K];
    __shared__ float rv[8];
    __shared__ int   rp[8];

    const int m = blockIdx.x;
    const int t = threadIdx.x;
    const int si = sidx[m];
    const float cx = points[3 * si + 0];
    const float cy = points[3 * si + 1];
    const float cz = points[3 * si + 2];

    float bd[KNN_K]; int bi[KNN_K];
#pragma unroll
    for (int s = 0; s < KNN_K; ++s) { bd[s] = FLT_BIG; bi[s] = 0; }
    float wvv = FLT_BIG; int wp = 0;

    for (int j = 0; j < N_POINTS / 256; ++j) {
        const int idx = t + 256 * j;
        const float dx = points[3 * idx + 0] - cx;
        const float dy = points[3 * idx + 1] - cy;
        const float dz = points[3 * idx + 2] - cz;
        const float d = dx * dx + dy * dy + dz * dz;
        if (d < wvv) {
#pragma unroll
            for (int s = 0; s < KNN_K; ++s)
                if (s == wp) { bd[s] = d; bi[s] = idx; }
            wvv = bd[0]; wp = 0;
#pragma unroll
            for (int s = 1; s < KNN_K; ++s)
                if (bd[s] > wvv) { wvv = bd[s]; wp = s; }
        }
    }
#pragma unroll
    for (int s = 0; s < KNN_K; ++s) { cd[t * KNN_K + s] = bd[s]; ci[t * KNN_K + s] = bi[s]; }
    __syncthreads();

    const int lane = t & 31, wid = t >> 5;
    for (int r = 0; r < KNN_K; ++r) {
        float mv = cd[t * KNN_K]; int mp = t * KNN_K;
#pragma unroll
        for (int s = 1; s < KNN_K; ++s) {
            const float v = cd[t * KNN_K + s];
            if (v < mv) { mv = v; mp = t * KNN_K + s; }
        }
        for (int off = 16; off > 0; off >>= 1) {
            const float ov = __shfl_xor(mv, off, 32);
            const int   op = __shfl_xor(mp, off, 32);
            if (ov < mv || (ov == mv && op < mp)) { mv = ov; mp = op; }
        }
        if (lane == 0) { rv[wid] = mv; rp[wid] = mp; }
        __syncthreads();
        if (t == 0) {
            float bv = rv[0]; int bp = rp[0];
            for (int w = 1; w < 8; ++w)
                if (rv[w] < bv || (rv[w] == bv && rp[w] < bp)) { bv = rv[w]; bp = rp[w]; }
            knn_idx[m * KNN_K + r] = ci[bp];
            cd[bp] = FLT_BIG;
        }
        __syncthreads();
    }
}

// ---------------------------------------------------------------------------
// Kernel 3: fp32 WMMA GEMM: Out[row, c] = sum_k A[row][k] * W[c][k]
//   mode 0: A rows gathered from `features` via knn index (layer-1 input)
//   mode 1: A rows = Asrc rows with BN1+ReLU applied on load; Out == Asrc
//           is safe (each block stages its 16 rows to LDS before storing).
// Block: 256 threads = 8 waves; each wave does two 16x16 output tiles,
// 64 chained V_WMMA_F32_16X16X4_F32 per tile over K=256.
// A-tile LDS rows padded to 260 floats -> conflict-free lane-strided b64 reads.
// ---------------------------------------------------------------------------
#define A_PAD 260

__global__ void gemm_wmma(const float* Asrc,
                          const int* __restrict__ gidx,
                          const float* __restrict__ W,
                          const float* __restrict__ mu,
                          const float* __restrict__ inv,
                          const float* __restrict__ gamma,
                          const float* __restrict__ beta,
                          float* Out, int mode) {
    __shared__ float Ash[16 * A_PAD];
    const int t = threadIdx.x;
    const int blk = blockIdx.x;

    // Stage 16 x 256 A tile (coalesced; channel = t)
    float gm = 0.f, bt = 0.f;
    if (mode == 1) {   // fold BN affine: y = x*(inv*gamma) + (beta - mu*inv*gamma)
        const float s = inv[t] * gamma[t];
        gm = s; bt = beta[t] - mu[t] * s;
    }
#pragma unroll
    for (int r = 0; r < 16; ++r) {
        const int row = blk * 16 + r;
        float v;
        if (mode == 0) {
            v = Asrc[(size_t)gidx[row] * CHANNELS + t];
        } else {
            v = Asrc[(size_t)row * CHANNELS + t];
            v = fmaxf(v * gm + bt, 0.0f);          // BN1 + ReLU
        }
        Ash[r * A_PAD + t] = v;
    }
    __syncthreads();

    const int lane = t & 31, wv = t >> 5;
    const int lr = lane & 15;        // row of A frag / col of B frag
    const int hi = lane >> 4;        // K-half select (ISA 32-bit 16x4 layout)
    const float* aBase = &Ash[lr * A_PAD + 2 * hi];

#pragma unroll
    for (int cti = 0; cti < 2; ++cti) {
        const int ct = wv + 8 * cti;                         // column tile 0..15
        const float* bBase = W + (size_t)(ct * 16 + lr) * CHANNELS + 2 * hi;
        v8f acc = {};
#pragma unroll 4
        for (int kk = 0; kk < CHANNELS; kk += 4) {
            const v2f a = *(const v2f*)(aBase + kk);
            const v2f b = *(const v2f*)(bBase + kk);
            acc = __builtin_amdgcn_wmma_f32_16x16x4_f32(
                false, a, false, b, (short)0, acc, false, false);
        }
        // C/D layout: VGPR r -> M = r + 8*hi, N = lr
        const int col = ct * 16 + lr;
#pragma unroll
        for (int r = 0; r < 8; ++r) {
            const int row = blk * 16 + r + 8 * hi;
            Out[(size_t)row * CHANNELS + col] = acc[r];
        }
    }
}

// ---------------------------------------------------------------------------
// BatchNorm statistics: deterministic two-stage column reduction (no atomics).
// ---------------------------------------------------------------------------
__global__ void stats_partial(const float* __restrict__ X,
                              float* __restrict__ psum, float* __restrict__ pssq) {
    const int t = threadIdx.x, blk = blockIdx.x;   // 128 blocks x 256 rows each
    const float* base = X + (size_t)blk * 256 * CHANNELS;
    float s = 0.f, q = 0.f;
    for (int r = 0; r < 256; ++r) {
        const float v = base[(size_t)r * CHANNELS + t];
        s += v; q += v * v;
    }
    psum[blk * CHANNELS + t] = s;
    pssq[blk * CHANNELS + t] = q;
}

__global__ void stats_final(const float* __restrict__ psum,
                            const float* __restrict__ pssq,
                            float* __restrict__ mu, float* __restrict__ inv) {
    const int t = threadIdx.x;                     // one block of 256 channels
    float s = 0.f, q = 0.f;
    for (int j = 0; j < MK / 256; ++j) { s += psum[j * CHANNELS + t]; q += pssq[j * CHANNELS + t]; }
    const float m = s * (1.0f / (float)MK);
    const float var = q * (1.0f / (float)MK) - m * m;   // biased, as jnp.var
    mu[t] = m;
    inv[t] = rsqrtf(var + BN_EPS);
}

// ---------------------------------------------------------------------------
// BN2 + ReLU + max-pool over K neighbors -> out_features [M, C]
// ---------------------------------------------------------------------------
__global__ void bn_relu_maxpool(const float* __restrict__ H,
                                const float* __restrict__ mu,
                                const float* __restrict__ inv,
                                const float* __restrict__ gamma,
                                const float* __restrict__ beta,
                                float* __restrict__ out) {
    const int m = blockIdx.x, t = threadIdx.x;
    const float g = inv[t] * gamma[t];
    const float b = beta[t] - mu[t] * g;
    float best = -FLT_BIG;
#pragma unroll
    for (int k = 0; k < KNN_K; ++k) {
        const float x = H[((size_t)m * KNN_K + k) * CHANNELS + t];
        best = fmaxf(best, fmaxf(x * g + b, 0.0f));
    }
    out[(size_t)m * CHANNELS + t] = best;
}

// ---------------------------------------------------------------------------
// Host-side orchestration (all on `stream`; graph-capture safe).
// Inputs: points, features, W1, gamma1, beta1, W2, gamma2, beta2 (all fp32)
// Output: [sampled_points (2048*3) | out_features (2048*256)] fp32
// ---------------------------------------------------------------------------
extern "C" void kernel_launch(void* const* d_in, const int* in_sizes, int n_in,
                              void* d_out, int out_size, void* d_ws, size_t ws_size,
                              hipStream_t stream) {
    const float* points   = (const float*)d_in[0];
    const float* features = (const float*)d_in[1];
    const float* W1       = (const float*)d_in[2];
    const float* gamma1   = (const float*)d_in[3];
    const float* beta1    = (const float*)d_in[4];
    const float* W2       = (const float*)d_in[5];
    const float* gamma2   = (const float*)d_in[6];
    const float* beta2    = (const float*)d_in[7];

    float* out = (float*)d_out;
    float* sampled_points = out;                       // [2048, 3]
    float* out_features   = out + NUM_SAMPLES * 3;     // [2048, 256]

    // Workspace layout (~34.3 MB)
    char* ws = (char*)d_ws;
    int*   sidx = (int*)ws;                                       // 2048 ints
    int*   knn  = (int*)(ws + 8192);                              // 32768 ints
    float* h1   = (float*)(ws + 139264);                          // 32768*256 f32 (reused in place)
    char*  p    = ws + 139264 + (size_t)MK * CHANNELS * 4;
    float* psum = (float*)p;  p += 128 * CHANNELS * 4;
    float* pssq = (float*)p;  p += 128 * CHANNELS * 4;
    float* mu1  = (float*)p;  p += CHANNELS * 4;
    float* inv1 = (float*)p;  p += CHANNELS * 4;
    float* mu2  = (float*)p;  p += CHANNELS * 4;
    float* inv2 = (float*)p;  p += CHANNELS * 4;

    // 1) FPS (one WGP; 192KB dynamic LDS for the point cloud)
    fps_kernel<<<1, 1024, 3 * N_POINTS * sizeof(float), stream>>>(points, sidx, sampled_points);

    // 2) KNN
    knn_kernel<<<NUM_SAMPLES, 256, 0, stream>>>(points, sidx, knn);

    // 3) Layer 1 GEMM (gather fused): h1 = gather(features) @ W1^T
    gemm_wmma<<<MK / 16, 256, 0, stream>>>(features, knn, W1,
                                           nullptr, nullptr, nullptr, nullptr, h1, 0);

    // 4) BN1 stats
    stats_partial<<<128, 256, 0, stream>>>(h1, psum, pssq);
    stats_final<<<1, 256, 0, stream>>>(psum, pssq, mu1, inv1);

    // 5) Layer 2 GEMM, BN1+ReLU fused on A-load, in place: h1 = relu(BN1(h1)) @ W2^T
    gemm_wmma<<<MK / 16, 256, 0, stream>>>(h1, nullptr, W2,
                                           mu1, inv1, gamma1, beta1, h1, 1);

    // 6) BN2 stats
    stats_partial<<<128, 256, 0, stream>>>(h1, psum, pssq);
    stats_final<<<1, 256, 0, stream>>>(psum, pssq, mu2, inv2);

    // 7) BN2 + ReLU + max-pool over K
    bn_relu_maxpool<<<NUM_SAMPLES, 256, 0, stream>>>(h1, mu2, inv2, gamma2, beta2, out_features);
}